// JzazbzShiftTorch_68298569941625
// MI455X (gfx1250) — compile-verified
//
#include <hip/hip_runtime.h>

// ---------------------------------------------------------------------------
// Jzazbz color shift, (32,3,512,512) f32 NCHW. Memory-bound streaming kernel:
//  - async global->LDS b128 staging (ASYNCcnt double-buffer pipeline)
//  - nontemporal b128 stores
//  - raw v_log_f32 / v_exp_f32 / v_rcp_f32 for all pow()/div()
// ---------------------------------------------------------------------------

typedef float v4f __attribute__((ext_vector_type(4)));
typedef int   v4i_gcc __attribute__((vector_size(16)));   // matches builtin param

#ifndef __has_builtin
#define __has_builtin(x) 0
#endif

#if __has_builtin(__builtin_amdgcn_global_load_async_to_lds_b128)
#define HAS_ASYNC_LDS 1
#else
#define HAS_ASYNC_LDS 0
#endif
#if __has_builtin(__builtin_amdgcn_s_wait_asynccnt)
#define HAS_WAIT_ASYNC 1
#else
#define HAS_WAIT_ASYNC 0
#endif

// global (addrspace 1) and LDS (addrspace 3) 16-byte-vector pointers
#define AS_G(p) ((__attribute__((address_space(1))) v4i_gcc*)(void*)(p))
#define AS_L(p) ((__attribute__((address_space(3))) v4i_gcc*)(void*)(p))

#define J_HW4     65536    // (512*512)/4 float4 per channel plane
#define J_PLANE3  196608   // 3*J_HW4 float4 per batch image
#define J_BLOCK   256
#define J_GRID    2048
#define J_ITERS   4
#define J_STRIDE  (J_BLOCK * J_GRID)   // 524288; *4 iters = 2^21 groups total

// ---------------- fast scalar math (hardware trans ops) --------------------

__device__ __forceinline__ float fpow_(float x, float p) {
  // x > 0 expected (callers clamp); pow(x,p) = exp2(p*log2(x))
  return __builtin_amdgcn_exp2f(p * __builtin_amdgcn_logf(x));
}
__device__ __forceinline__ float frcp_(float x) { return __builtin_amdgcn_rcpf(x); }

__device__ __forceinline__ float srgb2lin(float x) {
  float hi = fpow_((fmaxf(x, 0.0f) + 0.055f) * (1.0f / 1.055f), 2.4f);
  float lo = x * (1.0f / 12.92f);
  return (x <= 0.04045f) ? lo : hi;
}
__device__ __forceinline__ float lin2srgb(float v) {
  float hi = 1.055f * fpow_(fmaxf(v, 1e-12f), (1.0f / 2.4f)) - 0.055f;
  float lo = 12.92f * v;
  return (v <= 0.0031308f) ? lo : hi;
}
__device__ __forceinline__ float pq_(float x) {
  // t = (x/1e4)^N ; y = ((c1 + c2 t)/(1 + c3 t))^P
  float t = fpow_(fmaxf(x, 0.0f) * 1e-4f, 0.1593017578125f);
  return fpow_((0.8359375f + 18.8515625f * t) * frcp_(1.0f + 18.6875f * t),
               134.034375f);
}
__device__ __forceinline__ float pq_inv_(float y) {
  float yp = fpow_(fmaxf(y, 1e-12f), 7.460772656268214e-3f);
  float t  = (yp - 0.8359375f) * frcp_(18.8515625f - 18.6875f * yp);
  return 1e4f * fpow_(fmaxf(t, 0.0f), 6.277394636015326f);
}

__device__ __forceinline__ void jz_px(float r, float g, float b,
                                      float& ro, float& go, float& bo) {
  // sRGB decode ([-1,1] -> [0,1] -> linear)
  float lr = srgb2lin(fmaf(r, 0.5f, 0.5f));
  float lg = srgb2lin(fmaf(g, 0.5f, 0.5f));
  float lb = srgb2lin(fmaf(b, 0.5f, 0.5f));
  // sRGB -> XYZ
  float X = 0.4124564f * lr + 0.3575761f * lg + 0.1804375f * lb;
  float Y = 0.2126729f * lr + 0.7151522f * lg + 0.0721750f * lb;
  float Z = 0.0193339f * lr + 0.1191920f * lg + 0.9503041f * lb;
  // chromatic adaptation
  float Xp = 1.15f * X - 0.15f * Z;
  float Yp = 0.66f * Y + 0.34f * X;
  // M1
  float L =  0.41478972f * Xp + 0.579999f * Yp + 0.014648f  * Z;
  float M = -0.20151f    * Xp + 1.120649f * Yp + 0.0531008f * Z;
  float S = -0.0166008f  * Xp + 0.2648f   * Yp + 0.6684799f * Z;
  // PQ + M2
  float Lp = pq_(L), Mp = pq_(M), Sp = pq_(S);
  float Iz = 0.5f * (Lp + Mp);
  float az = 3.524f    * Lp - 4.066708f * Mp + 0.542708f * Sp + 0.001f;  // +DAZ
  float bz = 0.199076f * Lp + 1.096799f * Mp - 1.295875f * Sp - 0.001f;  // +DBZ
  // Jz nonlinearity + DJZ shift; D0 cancels exactly in jz+D0
  float jzd = 0.44f * Iz * frcp_(1.0f - 0.56f * Iz) + 0.002f;
  float Iz2 = jzd * frcp_(0.44f + 0.56f * jzd);
  // M2^-1
  float l2 = Iz2 + 0.1386050432715393f * az + 0.0580473161561189f * bz;
  float m2 = Iz2 - 0.1386050432715393f * az - 0.0580473161561189f * bz;
  float s2 = Iz2 - 0.0960192420263189f * az - 0.8118918961866040f * bz;
  // PQ^-1 + M1^-1
  float L2 = pq_inv_(l2), M2v = pq_inv_(m2), S2 = pq_inv_(s2);
  float xp =  1.9242264357876067f * L2 - 1.0047923125953657f * M2v + 0.0376514040306182f * S2;
  float yq =  0.3503167620949991f * L2 + 0.7264811939316552f * M2v - 0.0653844229480850f * S2;
  float z2 = -0.0909828109828475f * L2 - 0.3127282905230739f * M2v + 1.5227665613052603f * S2;
  float X2 = (xp + 0.15f * z2) * (1.0f / 1.15f);
  float Y2 = (yq - 0.34f * X2) * (1.0f / 0.66f);
  // XYZ -> sRGB
  float rl =  3.2404542f * X2 - 1.5371385f * Y2 - 0.4985314f * z2;
  float gl = -0.9692660f * X2 + 1.8760108f * Y2 + 0.0415560f * z2;
  float bl =  0.0556434f * X2 - 0.2040259f * Y2 + 1.0572252f * z2;
  ro = fmaf(lin2srgb(rl), 2.0f, -1.0f);
  go = fmaf(lin2srgb(gl), 2.0f, -1.0f);
  bo = fmaf(lin2srgb(bl), 2.0f, -1.0f);
}

// ---------------- async-count waits ----------------------------------------

__device__ __forceinline__ void wait_async_le3() {
#if HAS_WAIT_ASYNC
  __builtin_amdgcn_s_wait_asynccnt(3);
#else
  asm volatile("s_wait_asynccnt 3" ::: "memory");
#endif
}
__device__ __forceinline__ void wait_async_le0() {
#if HAS_WAIT_ASYNC
  __builtin_amdgcn_s_wait_asynccnt(0);
#else
  asm volatile("s_wait_asynccnt 0" ::: "memory");
#endif
}

// ---------------- kernel ----------------------------------------------------

__global__ __launch_bounds__(J_BLOCK) void jz_shift_kernel(
    const v4f* __restrict__ in, v4f* __restrict__ out) {
  const int tid = threadIdx.x;
  const int tg  = blockIdx.x * J_BLOCK + tid;

#if HAS_ASYNC_LDS
  // double-buffered per-thread staging: 2 stages x 3 planes x 16B = 96 B/thread
  __shared__ v4f stage[2][3][J_BLOCK];
  auto issue = [&](int k) {
    int    q    = tg + k * J_STRIDE;            // float4-group id, < 2^21
    int    bi   = q >> 16;                      // q / J_HW4  (batch)
    int    s4   = q & 0xFFFF;                   // spatial float4 index
    const v4f* gp = in + (size_t)bi * J_PLANE3 + s4;
    __builtin_amdgcn_global_load_async_to_lds_b128(AS_G(gp),
        AS_L(&stage[k & 1][0][tid]), 0, 0);
    __builtin_amdgcn_global_load_async_to_lds_b128(AS_G(gp + J_HW4),
        AS_L(&stage[k & 1][1][tid]), 0, 0);
    __builtin_amdgcn_global_load_async_to_lds_b128(AS_G(gp + 2 * J_HW4),
        AS_L(&stage[k & 1][2][tid]), 0, 0);
  };
  issue(0);
#endif

  for (int k = 0; k < J_ITERS; ++k) {
    int    q    = tg + k * J_STRIDE;
    int    bi   = q >> 16;
    int    s4   = q & 0xFFFF;
    size_t base = (size_t)bi * J_PLANE3 + s4;

    v4f R, G, B;
#if HAS_ASYNC_LDS
    if (k + 1 < J_ITERS) { issue(k + 1); wait_async_le3(); }
    else                 { wait_async_le0(); }
    asm volatile("" ::: "memory");   // don't hoist LDS reads above the wait
    R = stage[k & 1][0][tid];
    G = stage[k & 1][1][tid];
    B = stage[k & 1][2][tid];
#else
    R = __builtin_nontemporal_load(in + base);
    G = __builtin_nontemporal_load(in + base + J_HW4);
    B = __builtin_nontemporal_load(in + base + 2 * J_HW4);
#endif

    v4f Ro, Go, Bo;
#pragma unroll
    for (int i = 0; i < 4; ++i) {
      float ro, go, bo;
      jz_px(R[i], G[i], B[i], ro, go, bo);
      Ro[i] = ro; Go[i] = go; Bo[i] = bo;
    }

    __builtin_nontemporal_store(Ro, out + base);
    __builtin_nontemporal_store(Go, out + base + J_HW4);
    __builtin_nontemporal_store(Bo, out + base + 2 * J_HW4);
  }
}

// ---------------- host launch ----------------------------------------------

extern "C" void kernel_launch(void* const* d_in, const int* in_sizes, int n_in,
                              void* d_out, int out_size, void* d_ws, size_t ws_size,
                              hipStream_t stream) {
  (void)in_sizes; (void)n_in; (void)out_size; (void)d_ws; (void)ws_size;
  const v4f* in  = (const v4f*)d_in[0];
  v4f*       out = (v4f*)d_out;
  jz_shift_kernel<<<J_GRID, J_BLOCK, 0, stream>>>(in, out);
}